// SyntaxesAttention_85950885528509
// MI455X (gfx1250) — compile-verified
//
#include <hip/hip_runtime.h>
#include <hip/hip_bf16.h>

// SyntaxesAttention — exact algebraic reduction.
// All consumed attention rows collapse to v[b,h,0,:], so
//   y[b] = (inputs[b, 3968, :] @ Wv.T) @ Wo.T
//   out  = inputs, with rows [3968:4096) of each batch set to y[b].
// Memory-bound: ~128 MiB copy traffic (~5.5 us @ 23.3 TB/s); the two tiny
// fp32 GEMVs run on the tensor core via V_WMMA_F32_16X16X4_F32.

typedef float v2f __attribute__((ext_vector_type(2)));
typedef float v4f __attribute__((ext_vector_type(4)));
typedef float v8f __attribute__((ext_vector_type(8)));

#define HID  1024
#define SEQ  4096
#define NB   4
#define CS   3968   // context start row
#define CTXL 128    // rows overwritten per batch

// ---------------------------------------------------------------------------
// 1) Identity copy inputs -> out, float4-vectorized (HBM bound).
__global__ void sa_copy_inputs(const float* __restrict__ in,
                               float* __restrict__ out, int n4) {
    int i = blockIdx.x * blockDim.x + threadIdx.x;
    if (i < n4) {
        reinterpret_cast<v4f*>(out)[i] =
            reinterpret_cast<const v4f*>(in)[i];
    }
}

// ---------------------------------------------------------------------------
// 2/3) Out[m][n] = sum_k X[m*xstride + k] * W[n*HID + k],  m=0..3, n=0..1023.
// One wave (32 lanes) owns a 16-column tile of N; grid.x = HID/16 = 64.
// A-frag (16x4 f32): lane l, half=l>>4 -> X[l&15][k0+2*half .. +1]  (float2)
//   Lanes with m>=4 re-read rows m&3; their D rows are never consumed, so no
//   lane masking is needed and EXEC stays all-ones around the WMMA.
// B-frag (4x16 f32): W[n0+(l&15)][k0+2*half .. +1]                 (float2)
// D (16x16 f32, 8 VGPRs): rows 0..3 live in c[0..3] on lanes 0..15.
__global__ void sa_matvec4_wmma(const float* __restrict__ X, long xstride,
                                const float* __restrict__ W,
                                float* __restrict__ Out) {
    const int lane = threadIdx.x;      // 0..31
    const int half = lane >> 4;        // 0/1 selects K pair within the step
    const int lrow = lane & 15;
    const int n0   = blockIdx.x * 16;
    const int mrow = lrow & 3;         // fold 16 A-rows onto the 4 real rows

    const float* __restrict__ xrow = X + (long)mrow * xstride;
    const float* __restrict__ wrow = W + (long)(n0 + lrow) * HID;

    v8f c = {0.f, 0.f, 0.f, 0.f, 0.f, 0.f, 0.f, 0.f};

    #pragma unroll 4
    for (int k0 = 0; k0 < HID; k0 += 4) {
        const int k = k0 + 2 * half;
        v2f a = *reinterpret_cast<const v2f*>(xrow + k);
        v2f b = *reinterpret_cast<const v2f*>(wrow + k);
        // D = A(16x4) x B(4x16) + C, fp32 end-to-end (exact vs reference).
        c = __builtin_amdgcn_wmma_f32_16x16x4_f32(
                /*neg_a=*/false, a, /*neg_b=*/false, b,
                /*c_mod=*/(short)0, c, /*reuse_a=*/false, /*reuse_b=*/false);
    }

    if (half == 0) {  // lanes 0..15 hold D rows 0..3 in c[0..3]
        Out[0 * HID + n0 + lrow] = c[0];
        Out[1 * HID + n0 + lrow] = c[1];
        Out[2 * HID + n0 + lrow] = c[2];
        Out[3 * HID + n0 + lrow] = c[3];
    }
}

// ---------------------------------------------------------------------------
// 4) Broadcast Y[b] (1024 floats) across the 128 replaced rows per batch.
// 4 batches * 128 rows * 256 float4 = 131072 threads.
__global__ void sa_broadcast_rows(const float* __restrict__ Y,
                                  float* __restrict__ out) {
    int tid  = blockIdx.x * blockDim.x + threadIdx.x;   // 0..131071
    int rowg = tid >> 8;        // 0..511 : global (batch,row)
    int col4 = tid & 255;       // float4 column
    int b    = rowg >> 7;       // 0..3
    int r    = rowg & 127;      // 0..127
    v4f v = reinterpret_cast<const v4f*>(Y)[b * 256 + col4];
    reinterpret_cast<v4f*>(out)[((long)b * SEQ + CS + r) * 256 + col4] = v;
}

// ---------------------------------------------------------------------------
extern "C" void kernel_launch(void* const* d_in, const int* in_sizes, int n_in,
                              void* d_out, int out_size, void* d_ws, size_t ws_size,
                              hipStream_t stream) {
    // setup_inputs order: inputs, Wq, Wk, Wv, Wo, conv_w, conv_b (all fp32)
    const float* inputs = (const float*)d_in[0];
    const float* Wv     = (const float*)d_in[3];
    const float* Wo     = (const float*)d_in[4];
    float* out = (float*)d_out;
    float* ws  = (float*)d_ws;

    float* V = ws;            // 4 x 1024 fp32
    float* Y = ws + NB * HID; // 4 x 1024 fp32

    // 1) out = inputs (overwritten slice rewritten below; stream-ordered)
    const int n4 = NB * SEQ * HID / 4;  // 4,194,304 float4s
    sa_copy_inputs<<<n4 / 256, 256, 0, stream>>>(inputs, out, n4);

    // 2) V[b] = inputs[b, 3968, :] @ Wv.T   (X rows strided by batch stride)
    sa_matvec4_wmma<<<HID / 16, 32, 0, stream>>>(
        inputs + (long)CS * HID, (long)SEQ * HID, Wv, V);

    // 3) Y[b] = V[b] @ Wo.T
    sa_matvec4_wmma<<<HID / 16, 32, 0, stream>>>(V, (long)HID, Wo, Y);

    // 4) out[b, 3968+r, :] = Y[b]  for r in [0,128)
    sa_broadcast_rows<<<(NB * CTXL * (HID / 4)) / 256, 256, 0, stream>>>(Y, out);
}